// Pytorch3dRasterizer_6768868458599
// MI455X (gfx1250) — compile-verified
//
#include <hip/hip_runtime.h>
#include <math.h>
#include <stdint.h>

typedef __attribute__((ext_vector_type(2))) float v2f;
typedef __attribute__((ext_vector_type(8))) float v8f;

#define IMG 224
#define NB  2
#define NV  5023
#define NF  9976
#define ND  16
#define EPSD 1e-8f

#define CHUNK 128            // faces staged in LDS per pipeline stage
#define ROWS  (CHUNK * 3)    // float4 rows per stage
#define TILES_PER_BLOCK 8    // 8 waves * 16 pixels = 128 pixels per block
#define BLOCK_THREADS 256

// ---------------------------------------------------------------------------
// Pass 1: per-face affine coefficients (denominator folded in).
//   w0 = a0*px + b0*py + c0 ; w1 = a1*px + b1*py + c1 ; z = az*px + bz*py + cz
// Degenerate faces get rows (0,0,-1,0): w0 = -1 -> never valid.
// ---------------------------------------------------------------------------
__global__ void raster_coef_kernel(const float* __restrict__ verts,
                                   const int*   __restrict__ faces,
                                   float4*      __restrict__ coef) {
    int t = blockIdx.x * blockDim.x + threadIdx.x;
    if (t >= NB * NF) return;
    int n = t / NF;
    const int* fi = faces + (size_t)t * 3;
    int i0 = fi[0], i1 = fi[1], i2 = fi[2];
    const float* vb = verts + (size_t)n * NV * 3;
    // reference flips x,y: fixed = vertices * (-1,-1,+1)
    float x0 = -vb[i0*3+0], y0 = -vb[i0*3+1], z0 = vb[i0*3+2];
    float x1 = -vb[i1*3+0], y1 = -vb[i1*3+1], z1 = vb[i1*3+2];
    float x2 = -vb[i2*3+0], y2 = -vb[i2*3+1], z2 = vb[i2*3+2];

    float denom = (y1 - y2) * (x0 - x2) + (x2 - x1) * (y0 - y2);
    float4 r0, r1, rz;
    if (fabsf(denom) > EPSD) {
        float inv = 1.0f / denom;
        float a0 = (y1 - y2) * inv, b0 = (x2 - x1) * inv;
        float c0 = -a0 * x2 - b0 * y2;
        float a1 = (y2 - y0) * inv, b1 = (x0 - x2) * inv;
        float c1 = -a1 * x2 - b1 * y2;
        float dz0 = z0 - z2, dz1 = z1 - z2;
        r0 = make_float4(a0, b0, c0, 0.f);
        r1 = make_float4(a1, b1, c1, 0.f);
        rz = make_float4(dz0*a0 + dz1*a1,
                         dz0*b0 + dz1*b1,
                         dz0*c0 + dz1*c1 + z2, 0.f);
    } else {
        r0 = make_float4(0.f, 0.f, -1.f, 0.f);
        r1 = make_float4(0.f, 0.f, -1.f, 0.f);
        rz = make_float4(0.f, 0.f, -1.f, 0.f);
    }
    coef[(size_t)t * 3 + 0] = r0;
    coef[(size_t)t * 3 + 1] = r1;
    coef[(size_t)t * 3 + 2] = rz;
}

// ---------------------------------------------------------------------------
// Pass 2: WMMA rasterizer with double-buffered async global->LDS staging.
//   A (16 faces x K=4) = coefficient rows; B (K=4 x 16 pixels) = (px,py,1,0).
//   D layout: lane L -> pixel N = L&15, faces M = vgpr + (L>=16 ? 8 : 0).
// ---------------------------------------------------------------------------
__global__ void __launch_bounds__(BLOCK_THREADS)
raster_main_kernel(const float4* __restrict__ coef,
                   const float*  __restrict__ attrs,
                   float*        __restrict__ out) {
    __shared__ float4 sbuf[2][ROWS];   // 2 x 6 KB double buffer

    const int tid  = threadIdx.x;
    const int lane = tid & 31;
    const int wave = tid >> 5;
    const int half = lane >> 4;      // 0: K rows 0,1 / faces 0-7; 1: K rows 2,3 / faces 8-15
    const int p    = lane & 15;      // pixel within tile

    const int blocksPerImg = (IMG * IMG) / (16 * TILES_PER_BLOCK);  // 392
    const int n    = blockIdx.x / blocksPerImg;
    const int blk  = blockIdx.x % blocksPerImg;
    const int tile = blk * TILES_PER_BLOCK + wave;
    const int pixBase = tile * 16;
    const int h    = pixBase / IMG;
    const int wcol = pixBase % IMG;          // 224 = 14*16: tile stays in one row

    const float px = 1.f - (2.f * (float)(wcol + p) + 1.f) / (float)IMG;
    const float py = 1.f - (2.f * (float)h        + 1.f) / (float)IMG;
    // B operand: lanes 0-15 hold rows K=0 (px), K=1 (py); lanes 16-31 rows K=2 (1), K=3 (0)
    v2f B;
    B.x = half ? 1.0f : px;
    B.y = half ? 0.0f : py;

    const float4* cbase = coef + (size_t)n * NF * 3;

    // Stage CHUNK faces of coefficients into LDS buffer `b` via async DMA.
    // In-range rows: global_load_async_to_lds_b128 (ASYNCcnt, no VGPR round-trip).
    // Out-of-range tail rows: plain ds_store of an always-invalid row.
    auto stage = [&](int chunk, int b) {
        for (int i = tid; i < ROWS; i += BLOCK_THREADS) {
            int fl = i / 3;
            int fg = chunk + fl;
            if (fg < NF) {
                uint32_t ldsa = (uint32_t)(uintptr_t)&sbuf[b][i];
                const float4* gp = cbase + (size_t)fg * 3 + (i - fl * 3);
                asm volatile("global_load_async_to_lds_b128 %0, %1, off"
                             :: "v"(ldsa), "v"(gp) : "memory");
            } else {
                sbuf[b][i] = make_float4(0.f, 0.f, -1.f, 0.f);
            }
        }
    };

    stage(0, 0);
    asm volatile("s_wait_asynccnt 0x0" ::: "memory");
    __syncthreads();

    float bestz = INFINITY;
    int   bestf = -1;
    int   buf   = 0;

    for (int chunk = 0; chunk < NF; chunk += CHUNK) {
        if (chunk + CHUNK < NF) {
            // overlap next chunk's DMA with this chunk's WMMA work
            __builtin_prefetch((const char*)(cbase + (size_t)(chunk + 2 * CHUNK) * 3) + tid * 64, 0, 0);
            stage(chunk + CHUNK, buf ^ 1);
        }

        #pragma unroll 2
        for (int ft = 0; ft < CHUNK / 16; ++ft) {
            // A operands from LDS: lane = face (p), half selects coefficient pair (a,b)|(c,0)
            const float* srow = (const float*)&sbuf[buf][(ft * 16 + p) * 3];
            v2f A0 = *(const v2f*)(srow + 0 * 4 + half * 2);
            v2f A1 = *(const v2f*)(srow + 1 * 4 + half * 2);
            v2f Az = *(const v2f*)(srow + 2 * 4 + half * 2);

            v8f z8 = {};
            v8f W0 = __builtin_amdgcn_wmma_f32_16x16x4_f32(false, A0, false, B, (short)0, z8, false, false);
            v8f W1 = __builtin_amdgcn_wmma_f32_16x16x4_f32(false, A1, false, B, (short)0, z8, false, false);
            v8f Z  = __builtin_amdgcn_wmma_f32_16x16x4_f32(false, Az, false, B, (short)0, z8, false, false);

            const int fbase = chunk + ft * 16 + half * 8;
            #pragma unroll
            for (int e = 0; e < 8; ++e) {
                float w0 = W0[e], w1 = W1[e];
                float w2 = 1.f - w0 - w1;
                float z  = Z[e];
                // inside & front test as one min-reduce: all of {w0,w1,w2,z} >= 0
                float m  = fminf(fminf(w0, w1), fminf(w2, z));   // v_min3 path
                bool win = (m >= 0.f) && (z < bestz);  // strict <: keeps earliest face (argmin)
                bestf = win ? (fbase + e) : bestf;
                bestz = win ? z : bestz;
            }
        }

        asm volatile("s_wait_asynccnt 0x0" ::: "memory");
        __syncthreads();
        buf ^= 1;
    }

    // merge the two half-lanes covering the same pixel (faces 0-7 vs 8-15 of each tile)
    float oz = __shfl_xor(bestz, 16, 32);
    int   of = __shfl_xor(bestf, 16, 32);
    bool otherWins = (of >= 0) &&
                     ((bestf < 0) || (oz < bestz) || ((oz == bestz) && (of < bestf)));
    bestf = otherWins ? of : bestf;

    // epilogue: recompute winning barycentrics from L2-resident coefficients,
    // each half-lane writes 8 of the 16 attribute channels
    const size_t outImg = (size_t)n * 17 * IMG * IMG;
    const size_t pixOff = (size_t)h * IMG + (wcol + p);

    if (bestf >= 0) {
        float4 r0 = cbase[(size_t)bestf * 3 + 0];
        float4 r1 = cbase[(size_t)bestf * 3 + 1];
        float bw0 = r0.x * px + r0.y * py + r0.z;
        float bw1 = r1.x * px + r1.y * py + r1.z;
        float bw2 = 1.f - bw0 - bw1;
        const float* ab = attrs + ((size_t)n * NF + bestf) * 3 * ND;
        #pragma unroll
        for (int d = 0; d < 8; ++d) {
            int dd = half * 8 + d;
            float v = bw0 * ab[0 * ND + dd] + bw1 * ab[1 * ND + dd] + bw2 * ab[2 * ND + dd];
            out[outImg + (size_t)dd * IMG * IMG + pixOff] = v;
        }
    } else {
        #pragma unroll
        for (int d = 0; d < 8; ++d) {
            int dd = half * 8 + d;
            out[outImg + (size_t)dd * IMG * IMG + pixOff] = 0.f;
        }
    }
    if (half == 0)
        out[outImg + (size_t)16 * IMG * IMG + pixOff] = (bestf >= 0) ? 1.f : 0.f;
}

// ---------------------------------------------------------------------------
extern "C" void kernel_launch(void* const* d_in, const int* in_sizes, int n_in,
                              void* d_out, int out_size, void* d_ws, size_t ws_size,
                              hipStream_t stream) {
    const float* verts = (const float*)d_in[0];   // [2,5023,3] fp32
    const int*   faces = (const int*)  d_in[1];   // [2,9976,3] int32
    const float* attrs = (const float*)d_in[2];   // [2,9976,3,16] fp32
    float*       out   = (float*)d_out;           // [2,17,224,224] fp32
    float4*      coef  = (float4*)d_ws;           // 2*9976*3 float4 = ~958 KB

    int nFaces = NB * NF;
    raster_coef_kernel<<<(nFaces + 255) / 256, 256, 0, stream>>>(verts, faces, coef);

    int blocks = NB * ((IMG * IMG) / (16 * TILES_PER_BLOCK));  // 784
    raster_main_kernel<<<blocks, BLOCK_THREADS, 0, stream>>>(coef, attrs, out);
}